// get_first_edge_feature_9723805958423
// MI455X (gfx1250) — compile-verified
//
#include <hip/hip_runtime.h>

typedef float v2f __attribute__((ext_vector_type(2)));
typedef float v8f __attribute__((ext_vector_type(8)));

#define EPS   1e-3f
#define NPTS  4096
#define BATCH 4
#define KNN   20

__device__ __forceinline__ bool kless(float v1, int i1, float v2, int i2) {
    // smaller value first; ties broken by smaller index (matches jax top_k of -adj)
    return (v1 < v2) || (v1 == v2 && i1 < i2);
}

// gfx1250 async copy: global -> LDS, 16 bytes per lane, tracked by ASYNCcnt.
__device__ __forceinline__ void async_ld_b128(unsigned lds_byte_off, const void* g) {
    asm volatile("global_load_async_to_lds_b128 %0, %1, off"
                 :: "v"(lds_byte_off), "v"(g)
                 : "memory");
}
__device__ __forceinline__ void wait_async0() {
    asm volatile("s_wait_asynccnt 0x0" ::: "memory");
}

// ---------------------------------------------------------------------------
// Single fused kernel: per 16-row tile (256 threads = 8 waves):
//   phase 0: async-load x tile + W1 + W2 into LDS
//   phase 1: h1 = BN(ReLU(x@W1+b1))        (scalar, 16x32)
//   phase 2: h2 = BN(ReLU(h1@W2+b2))       (scalar, 16x64)
//   phase 3: h3 = BN(ReLU(h2@W3+b3))       (WMMA f32, 16x512 -> LDS)
//   phase 4: adj = h3@Wd + bd, 32 rounds of 128 cols, WMMA f32,
//            streaming per-row top-20-smallest (register lists)
//   phase 5: 16-way merge of sorted candidate lists, edge-feature gather
// adj (268 MB) and h3 (33 MB) never touch HBM.
// ---------------------------------------------------------------------------
__global__ __launch_bounds__(256) void edge_feature_fused_kernel(
    const float* __restrict__ x,
    const float* __restrict__ W1, const float* __restrict__ b1,
    const float* __restrict__ g1, const float* __restrict__ be1,
    const float* __restrict__ m1, const float* __restrict__ v1,
    const float* __restrict__ W2, const float* __restrict__ b2,
    const float* __restrict__ g2, const float* __restrict__ be2,
    const float* __restrict__ m2, const float* __restrict__ v2,
    const float* __restrict__ W3, const float* __restrict__ b3,
    const float* __restrict__ g3, const float* __restrict__ be3,
    const float* __restrict__ m3, const float* __restrict__ v3,
    const float* __restrict__ Wd, const float* __restrict__ bd,
    float* __restrict__ out)
{
    // LDS layout (byte offsets), phases ordered by barriers:
    //   [0,     33280)  W1 (2KB) + W2 (8KB)  ->  h3 tile float[16][520]
    //                   -> candidates: val[16][320] @0, idx[16][320] @20480
    //   [33280, 41728)  h1 float[16*33] + h2 float[16*66]  ->  adj chunk float[16][132]
    //   [41728, 42752)  x tile float[16][16]
    //   [42752, 44032)  topk idx int[16][20]
    __shared__ __align__(16) unsigned char smem[44032];
    float* w1s      = (float*)smem;                   // 512 f
    float* w2s      = (float*)(smem + 2048);          // 2048 f
    float* h3s      = (float*)smem;                   // 16 x 520 f
    float* h1s      = (float*)(smem + 33280);         // 16 x 33 f
    float* h2s      = (float*)(smem + 35392);         // 16 x 66 f
    float* adjc     = (float*)(smem + 33280);         // 16 x 132 f
    float* xt       = (float*)(smem + 41728);         // 16 x 16 f
    int*   topk     = (int*)(smem + 42752);           // 16 x 20 i
    float* cand_val = (float*)smem;                   // 16 x 320 f
    int*   cand_idx = (int*)(smem + 20480);           // 16 x 320 i

    const unsigned XT_OFF = 41728u, W1_OFF = 0u, W2_OFF = 2048u;

    const int tid  = threadIdx.x;
    const int lane = tid & 31;
    const int l15  = lane & 15;
    const int hi   = lane >> 4;
    const int wave = tid >> 5;                 // 0..7
    const int rowbase = blockIdx.x * 16;
    const int b = rowbase >> 12;               // 4096 rows per batch, tiles never straddle

    // ---- phase 0: async global->LDS: x tile (1KB) + W1 (2KB) + W2 (8KB) ----
    for (int j = tid; j < 704; j += 256) {
        if (j < 64) {
            async_ld_b128(XT_OFF + (unsigned)j * 16u,
                          (const char*)x + (size_t)rowbase * 64 + (size_t)j * 16);
        } else if (j < 192) {
            const int jj = j - 64;
            async_ld_b128(W1_OFF + (unsigned)jj * 16u, (const char*)W1 + (size_t)jj * 16);
        } else {
            const int jj = j - 192;
            async_ld_b128(W2_OFF + (unsigned)jj * 16u, (const char*)W2 + (size_t)jj * 16);
        }
    }
    wait_async0();
    __syncthreads();

    // ---- phase 1: layer 1 (16x16 @ 16x32), 2 outputs/thread ----
    for (int o = tid; o < 512; o += 256) {
        const int r = o >> 5, c = o & 31;
        float s = b1[c];
        #pragma unroll
        for (int d = 0; d < 16; ++d) s += xt[r * 16 + d] * w1s[d * 32 + c];
        s = fmaxf(s, 0.f);
        s = (s - m1[c]) * rsqrtf(v1[c] + EPS) * g1[c] + be1[c];
        h1s[r * 33 + c] = s;
    }
    __syncthreads();

    // ---- phase 2: layer 2 (16x32 @ 32x64), 4 outputs/thread ----
    for (int o = tid; o < 1024; o += 256) {
        const int r = o >> 6, c = o & 63;
        float s = b2[c];
        #pragma unroll
        for (int j = 0; j < 32; ++j) s += h1s[r * 33 + j] * w2s[j * 64 + c];
        s = fmaxf(s, 0.f);
        s = (s - m2[c]) * rsqrtf(v2[c] + EPS) * g2[c] + be2[c];
        h2s[r * 66 + c] = s;
    }
    __syncthreads();   // w1s/w2s dead from here; h3s region free to write

    // ---- phase 3: layer 3 (16x64 @ 64x512) via WMMA f32, result -> LDS ----
    for (int t = 0; t < 4; ++t) {
        const int ct  = wave * 4 + t;              // 32 column tiles / 8 waves
        const int col = ct * 16 + l15;
        v8f acc = {};
        #pragma unroll
        for (int kk = 0; kk < 16; ++kk) {
            const int k0 = kk * 4 + 2 * hi;        // A layout: lo lanes K={0,1}, hi K={2,3}
            v2f a = *(const v2f*)(&h2s[l15 * 66 + k0]);
            v2f bf;
            bf.x = W3[(k0 + 0) * 512 + col];
            bf.y = W3[(k0 + 1) * 512 + col];
            acc = __builtin_amdgcn_wmma_f32_16x16x4_f32(
                false, a, false, bf, (short)0, acc, false, false);
        }
        const float bb = b3[col];
        const float rs = rsqrtf(v3[col] + EPS) * g3[col];
        const float mm = m3[col], bt = be3[col];
        #pragma unroll
        for (int i = 0; i < 8; ++i) {
            float v = fmaxf(acc[i] + bb, 0.f);
            v = (v - mm) * rs + bt;
            h3s[(i + 8 * hi) * 520 + col] = v;     // C layout: vgpr i -> row i (+8 hi)
        }
    }
    __syncthreads();   // h1s/h2s dead; adjc region free

    // ---- phase 4: adj GEMM + streaming top-k ----
    float tv[KNN];
    int   ti[KNN];
    #pragma unroll
    for (int j = 0; j < KNN; ++j) { tv[j] = 3.0e38f; ti[j] = 0x7fffffff; }

    const int myrow = 2 * wave + hi;           // row owned by this half-wave
    const int colc  = wave * 16 + l15;         // column within 128-wide chunk

    for (int rd = 0; rd < 32; ++rd) {
        const int nbase = rd * 128;
        const int col = nbase + colc;

        // 16x16 tile, K = 512 (128 WMMAs), Wd streamed from L2
        v8f acc = {};
        #pragma unroll 8
        for (int kk = 0; kk < 128; ++kk) {
            const int k0 = kk * 4 + 2 * hi;
            v2f a = *(const v2f*)(&h3s[l15 * 520 + k0]);
            v2f bf;
            bf.x = Wd[(size_t)(k0 + 0) * 4096 + col];
            bf.y = Wd[(size_t)(k0 + 1) * 4096 + col];
            acc = __builtin_amdgcn_wmma_f32_16x16x4_f32(
                false, a, false, bf, (short)0, acc, false, false);
        }
        const float bias = bd[col];
        #pragma unroll
        for (int i = 0; i < 8; ++i)
            adjc[(i + 8 * hi) * 132 + colc] = acc[i] + bias;
        __syncthreads();

        // streaming top-k: each half-wave owns one row, 8 values/lane
        #pragma unroll
        for (int t = 0; t < 8; ++t) {
            const int cc = l15 + t * 16;
            const float v = adjc[myrow * 132 + cc];
            const int  id = nbase + cc;
            if (kless(v, id, tv[KNN - 1], ti[KNN - 1])) {   // threshold reject: ~1 cmp/value
                float cv = v; int ci = id;
                #pragma unroll
                for (int j = 0; j < KNN; ++j) {             // unrolled sorted insert
                    if (kless(cv, ci, tv[j], ti[j])) {
                        float tf = tv[j]; tv[j] = cv; cv = tf;
                        int   tn = ti[j]; ti[j] = ci; ci = tn;
                    }
                }
            }
        }
        __syncthreads();   // before next round overwrites adjc
    }

    // ---- phase 5: dump sorted lists, 16-way merge, edge-feature gather ----
    #pragma unroll
    for (int j = 0; j < KNN; ++j) {
        cand_val[myrow * 320 + l15 * KNN + j] = tv[j];
        cand_idx[myrow * 320 + l15 * KNN + j] = ti[j];
    }
    __syncthreads();

    if (tid < 16) {
        const int r = tid;
        int ptr[16];
        for (int h = 0; h < 16; ++h) ptr[h] = 0;
        for (int it = 0; it < KNN; ++it) {
            float best = 3.3e38f; int besti = 0x7fffffff; int bh = 0;
            for (int h = 0; h < 16; ++h) {
                if (ptr[h] < KNN) {
                    const float cv = cand_val[r * 320 + h * KNN + ptr[h]];
                    const int   ci = cand_idx[r * 320 + h * KNN + ptr[h]];
                    if (kless(cv, ci, best, besti)) { best = cv; besti = ci; bh = h; }
                }
            }
            topk[r * KNN + it] = besti;
            ptr[bh]++;
        }
    }
    __syncthreads();

    // out[b,n,k,0:16] = x[b,n]; out[b,n,k,16:32] = x[b,idx] - x[b,n]
    for (int p = tid; p < 16 * KNN; p += 256) {
        const int r = p / KNN, k = p % KNN;
        const int idx = topk[r * KNN + k];
        const float4* xc = (const float4*)(xt + r * 16);
        const float4* xn = (const float4*)(x + ((size_t)(b << 12) + idx) * 16);
        float4* ob = (float4*)(out + ((size_t)(rowbase + r) * KNN + k) * 32);
        #pragma unroll
        for (int q = 0; q < 4; ++q) {
            const float4 c = xc[q];
            const float4 n = xn[q];
            ob[q] = c;
            float4 d;
            d.x = n.x - c.x; d.y = n.y - c.y; d.z = n.z - c.z; d.w = n.w - c.w;
            ob[4 + q] = d;
        }
    }
}

// ---------------------------------------------------------------------------
extern "C" void kernel_launch(void* const* d_in, const int* in_sizes, int n_in,
                              void* d_out, int out_size, void* d_ws, size_t ws_size,
                              hipStream_t stream)
{
    const float* x   = (const float*)d_in[0];
    const float* W1  = (const float*)d_in[1];
    const float* b1  = (const float*)d_in[2];
    const float* g1  = (const float*)d_in[3];
    const float* be1 = (const float*)d_in[4];
    const float* m1  = (const float*)d_in[5];
    const float* v1  = (const float*)d_in[6];
    const float* W2  = (const float*)d_in[7];
    const float* b2  = (const float*)d_in[8];
    const float* g2  = (const float*)d_in[9];
    const float* be2 = (const float*)d_in[10];
    const float* m2  = (const float*)d_in[11];
    const float* v2  = (const float*)d_in[12];
    const float* W3  = (const float*)d_in[13];
    const float* b3  = (const float*)d_in[14];
    const float* g3  = (const float*)d_in[15];
    const float* be3 = (const float*)d_in[16];
    const float* m3  = (const float*)d_in[17];
    const float* v3  = (const float*)d_in[18];
    const float* Wd  = (const float*)d_in[19];
    const float* bd  = (const float*)d_in[20];

    float* out = (float*)d_out;
    (void)d_ws; (void)ws_size; (void)in_sizes; (void)n_in; (void)out_size;

    const int nblocks = (BATCH * NPTS) / 16;   // 1024

    edge_feature_fused_kernel<<<nblocks, 256, 0, stream>>>(
        x, W1, b1, g1, be1, m1, v1,
        W2, b2, g2, be2, m2, v2,
        W3, b3, g3, be3, m3, v3,
        Wd, bd, out);
}